// FlashNeoxAttention_33904471834715
// MI455X (gfx1250) — compile-verified
//
#include <hip/hip_runtime.h>
#include <hip/hip_bf16.h>

// ---------------------------------------------------------------------------
// FlashNeoxAttention for MI455X (gfx1250, wave32, WMMA).
// fp32 end-to-end using V_WMMA_F32_16X16X4_F32 (fp32 reference -> fp32 matrix
// pipe). ~206 GFLOP, all operands L2-resident. GEMM waves use 32x128 tiles
// (2 A-frags amortize the 8 B-frag loads over 16 WMMAs -> L2 traffic ~halved).
// ---------------------------------------------------------------------------

typedef float v2f __attribute__((ext_vector_type(2)));
typedef float v8f __attribute__((ext_vector_type(8)));

#define BB   2
#define SS   2048
#define HH   16
#define DD   128
#define HID  2048
#define NQKV 6144
#define TT   4096

__device__ __forceinline__ v8f wmma4(v2f a, v2f b, v8f c) {
    // D = A(16x4 f32) * B(4x16 f32) + C(16x16 f32)
    return __builtin_amdgcn_wmma_f32_16x16x4_f32(
        /*neg_a=*/false, a, /*neg_b=*/false, b,
        /*c_mod=*/(short)0, c, /*reuse_a=*/false, /*reuse_b=*/false);
}

// ---------------------------------------------------------------------------
// Kernel 1: qkv = hidden @ w_qkv^T + b_qkv ; RoPE on q,k ; scatter to
//   Q[B,H,S,D], K[B,H,S,D], Vt[B,H,D,S].
// One wave -> 32 tokens x 128 cols (= one head of one of q/k/v).
// ---------------------------------------------------------------------------
__global__ void __launch_bounds__(256)
qkv_rope_kernel(const float* __restrict__ hs,
                const float* __restrict__ cosp,
                const float* __restrict__ sinp,
                const float* __restrict__ wq,
                const float* __restrict__ bq,
                float* __restrict__ Q,
                float* __restrict__ K,
                float* __restrict__ Vt) {
    const int lane  = threadIdx.x & 31;
    const int wave  = blockIdx.x * (blockDim.x >> 5) + (threadIdx.x >> 5);
    const int mtile = wave / (NQKV / 128);   // 0..127 (32-token tile)
    const int nblk  = wave % (NQKV / 128);   // 0..47  (which,head)
    const int which = nblk >> 4;             // 0=q 1=k 2=v
    const int h     = nblk & 15;
    const int ln16  = lane & 15;
    const int lhalf = lane >> 4;

    v8f acc[2][8];
#pragma unroll
    for (int mt = 0; mt < 2; ++mt)
#pragma unroll
        for (int nt = 0; nt < 8; ++nt) acc[mt][nt] = (v8f){};

    const float* aptr0 = hs + (mtile * 32 + ln16) * HID + 2 * lhalf;
    const float* aptr1 = aptr0 + 16 * HID;
    const float* bbase = wq + (nblk * 128 + ln16) * HID + 2 * lhalf;

    for (int kk = 0; kk < HID; kk += 4) {
        if ((kk & 63) == 0 && kk + 64 < HID) {
            __builtin_prefetch(aptr0 + kk + 64, 0, 0);
            __builtin_prefetch(bbase + kk + 64, 0, 0);
        }
        v2f a0 = *(const v2f*)(aptr0 + kk);
        v2f a1 = *(const v2f*)(aptr1 + kk);
#pragma unroll
        for (int nt = 0; nt < 8; ++nt) {
            v2f b = *(const v2f*)(bbase + nt * 16 * HID + kk);
            acc[0][nt] = wmma4(a0, b, acc[0][nt]);
            acc[1][nt] = wmma4(a1, b, acc[1][nt]);
        }
    }

    // bias (same per column across the 8 rows of each accum VGPR)
#pragma unroll
    for (int nt = 0; nt < 8; ++nt) {
        const float bias = bq[nblk * 128 + nt * 16 + ln16];
#pragma unroll
        for (int mt = 0; mt < 2; ++mt)
#pragma unroll
            for (int i = 0; i < 8; ++i) acc[mt][nt][i] += bias;
    }

    const int rowoff = lhalf * 8;
#pragma unroll
    for (int mt = 0; mt < 2; ++mt) {
        const int tbase = mtile * 32 + mt * 16;
        if (which < 2) {
            float* dst = (which == 0) ? Q : K;
#pragma unroll
            for (int nt = 0; nt < 4; ++nt) {
                const int dcol = nt * 16 + ln16;            // 0..63
#pragma unroll
                for (int i = 0; i < 8; ++i) {
                    const int t = tbase + i + rowoff;       // global token
                    const float c = cosp[t * 64 + dcol];
                    const float s = sinp[t * 64 + dcol];
                    const float x1 = acc[mt][nt][i];
                    const float x2 = acc[mt][nt + 4][i];
                    const int b    = t >> 11;               // t / S
                    const int srow = t & (SS - 1);
                    const int base = ((b * HH + h) * SS + srow) * DD;
                    dst[base + dcol]      = x1 * c - x2 * s;
                    dst[base + dcol + 64] = x2 * c + x1 * s;
                }
            }
        } else {
#pragma unroll
            for (int nt = 0; nt < 8; ++nt) {
                const int d = nt * 16 + ln16;
#pragma unroll
                for (int i = 0; i < 8; ++i) {
                    const int t    = tbase + i + rowoff;
                    const int b    = t >> 11;
                    const int srow = t & (SS - 1);
                    Vt[((b * HH + h) * DD + d) * SS + srow] = acc[mt][nt][i];
                }
            }
        }
    }
}

// ---------------------------------------------------------------------------
// Kernel 2: causal flash attention. One wave per (b,h, 16-query tile).
// Scores via WMMA (Q frags cached, pre-scaled); online softmax in registers
// (shfl_xor reductions within 16-lane halves = C-matrix N striping);
// P converted C-layout -> A-layout through a per-wave LDS tile; P*V via WMMA.
// ---------------------------------------------------------------------------
__global__ void __launch_bounds__(256)
attn_kernel(const float* __restrict__ Q,
            const float* __restrict__ K,
            const float* __restrict__ Vt,
            float* __restrict__ ctx) {
    __shared__ float plds[8][16 * 16];   // one 16x16 P tile per wave
    const int lane  = threadIdx.x & 31;
    const int wib   = threadIdx.x >> 5;
    const int wave  = blockIdx.x * 8 + wib;
    const int qt    = wave & (SS / 16 - 1);   // 0..127
    const int bh    = wave >> 7;              // 0..31
    const int ln16  = lane & 15;
    const int lhalf = lane >> 4;

    // Q fragments, pre-scaled by D^-1/2 (folds softmax scale into QK^T)
    const float scale = 0.08838834764831845f;   // 128^-0.5
    const float* qbase = Q + (bh * SS + qt * 16 + ln16) * DD + 2 * lhalf;
    v2f qf[32];
#pragma unroll
    for (int kc = 0; kc < 32; ++kc) {
        v2f t = *(const v2f*)(qbase + kc * 4);
        t[0] *= scale; t[1] *= scale;
        qf[kc] = t;
    }

    float m_i[8], l_i[8];
    v8f o[8];
#pragma unroll
    for (int i = 0; i < 8; ++i) { m_i[i] = -1e30f; l_i[i] = 0.0f; }
#pragma unroll
    for (int nt = 0; nt < 8; ++nt) o[nt] = (v8f){};

    const float* kbase = K  + bh * SS * DD;
    const float* vbase = Vt + bh * DD * SS;
    float* pl = plds[wib];

    for (int kt = 0; kt <= qt; ++kt) {
        // ---- S = (Q*scale) @ K^T  (16x16 tile, K-dim = 128) ----
        v8f sc = (v8f){};
        const float* kptr = kbase + (kt * 16 + ln16) * DD + 2 * lhalf;
#pragma unroll
        for (int kc = 0; kc < 32; ++kc) {
            v2f kf = *(const v2f*)(kptr + kc * 4);
            sc = wmma4(qf[kc], kf, sc);
        }

        // ---- causal mask on the diagonal tile ----
        if (kt == qt) {
#pragma unroll
            for (int i = 0; i < 8; ++i) {
                const int row = i + lhalf * 8;
                if (ln16 > row) sc[i] = -1e30f;
            }
        }

        // ---- online softmax (row stats within 16-lane halves) ----
        float alpha[8];
#pragma unroll
        for (int i = 0; i < 8; ++i) {
            float r = sc[i];
            r = fmaxf(r, __shfl_xor(r, 1));
            r = fmaxf(r, __shfl_xor(r, 2));
            r = fmaxf(r, __shfl_xor(r, 4));
            r = fmaxf(r, __shfl_xor(r, 8));
            const float mn = fmaxf(m_i[i], r);
            alpha[i] = __expf(m_i[i] - mn);
            m_i[i] = mn;
            float p = __expf(sc[i] - mn);
            sc[i] = p;
            float rs = p;
            rs += __shfl_xor(rs, 1);
            rs += __shfl_xor(rs, 2);
            rs += __shfl_xor(rs, 4);
            rs += __shfl_xor(rs, 8);
            l_i[i] = l_i[i] * alpha[i] + rs;
        }
#pragma unroll
        for (int nt = 0; nt < 8; ++nt)
#pragma unroll
            for (int i = 0; i < 8; ++i) o[nt][i] *= alpha[i];

        // ---- P: C-layout -> LDS -> A-layout (per-wave tile, DS in-order) ----
#pragma unroll
        for (int i = 0; i < 8; ++i)
            pl[(i + lhalf * 8) * 16 + ln16] = sc[i];

        // ---- O += P @ V  (Vt layout makes B-frags contiguous b64 loads) ----
#pragma unroll
        for (int kc2 = 0; kc2 < 4; ++kc2) {
            v2f pa = *(const v2f*)(pl + ln16 * 16 + kc2 * 4 + 2 * lhalf);
            const float* vp = vbase + ln16 * SS + kt * 16 + kc2 * 4 + 2 * lhalf;
#pragma unroll
            for (int nt = 0; nt < 8; ++nt) {
                v2f vf = *(const v2f*)(vp + nt * 16 * SS);
                o[nt] = wmma4(pa, vf, o[nt]);
            }
        }
    }

    // ---- normalize and write ctx[T, HID] (t-major, h*128+d cols) ----
    const int b = bh >> 4, h = bh & 15;
#pragma unroll
    for (int i = 0; i < 8; ++i) {
        const float inv = 1.0f / l_i[i];
        const int t = b * SS + qt * 16 + i + lhalf * 8;
#pragma unroll
        for (int nt = 0; nt < 8; ++nt)
            ctx[t * HID + h * DD + nt * 16 + ln16] = o[nt][i] * inv;
    }
}

// ---------------------------------------------------------------------------
// Kernel 3: out = ctx @ w_dense^T + b_dense.  One wave -> 32x128 tile.
// ---------------------------------------------------------------------------
__global__ void __launch_bounds__(256)
dense_kernel(const float* __restrict__ ctx,
             const float* __restrict__ wd,
             const float* __restrict__ bd,
             float* __restrict__ out) {
    const int lane  = threadIdx.x & 31;
    const int wave  = blockIdx.x * (blockDim.x >> 5) + (threadIdx.x >> 5);
    const int mtile = wave / (HID / 128);   // 0..127
    const int nblk  = wave % (HID / 128);   // 0..15
    const int ln16  = lane & 15;
    const int lhalf = lane >> 4;

    v8f acc[2][8];
#pragma unroll
    for (int mt = 0; mt < 2; ++mt)
#pragma unroll
        for (int nt = 0; nt < 8; ++nt) acc[mt][nt] = (v8f){};

    const float* aptr0 = ctx + (mtile * 32 + ln16) * HID + 2 * lhalf;
    const float* aptr1 = aptr0 + 16 * HID;
    const float* bbase = wd + (nblk * 128 + ln16) * HID + 2 * lhalf;

    for (int kk = 0; kk < HID; kk += 4) {
        if ((kk & 63) == 0 && kk + 64 < HID) {
            __builtin_prefetch(aptr0 + kk + 64, 0, 0);
            __builtin_prefetch(bbase + kk + 64, 0, 0);
        }
        v2f a0 = *(const v2f*)(aptr0 + kk);
        v2f a1 = *(const v2f*)(aptr1 + kk);
#pragma unroll
        for (int nt = 0; nt < 8; ++nt) {
            v2f b = *(const v2f*)(bbase + nt * 16 * HID + kk);
            acc[0][nt] = wmma4(a0, b, acc[0][nt]);
            acc[1][nt] = wmma4(a1, b, acc[1][nt]);
        }
    }

#pragma unroll
    for (int nt = 0; nt < 8; ++nt) {
        const int n = nblk * 128 + nt * 16 + ln16;
        const float bias = bd[n];
#pragma unroll
        for (int mt = 0; mt < 2; ++mt)
#pragma unroll
            for (int i = 0; i < 8; ++i) {
                const int t = mtile * 32 + mt * 16 + i + lhalf * 8;
                out[t * HID + n] = acc[mt][nt][i] + bias;
            }
    }
}

// ---------------------------------------------------------------------------
extern "C" void kernel_launch(void* const* d_in, const int* in_sizes, int n_in,
                              void* d_out, int out_size, void* d_ws, size_t ws_size,
                              hipStream_t stream) {
    (void)in_sizes; (void)n_in; (void)out_size; (void)ws_size;
    const float* hs   = (const float*)d_in[0];   // [T, HID]
    const float* cosp = (const float*)d_in[1];   // [T, 64]
    const float* sinp = (const float*)d_in[2];   // [T, 64]
    const float* wq   = (const float*)d_in[3];   // [3*HID, HID]
    const float* bq   = (const float*)d_in[4];   // [3*HID]
    const float* wd   = (const float*)d_in[5];   // [HID, HID]
    const float* bd   = (const float*)d_in[6];   // [HID]
    float* out = (float*)d_out;

    // workspace: Q, K, Vt, ctx = 4 * 32 MB = 128 MB
    const size_t PER = (size_t)BB * HH * SS * DD;   // 8,388,608 floats
    float* Q   = (float*)d_ws;
    float* K   = Q + PER;
    float* Vt  = K + PER;
    float* ctx = Vt + PER;

    // K1: 128 token-tiles * 48 (which,head) blocks = 6144 waves / 8 per block
    qkv_rope_kernel<<<768, 256, 0, stream>>>(hs, cosp, sinp, wq, bq, Q, K, Vt);
    // K2: 32 (b,h) * 128 q-tiles = 4096 waves / 8 per block
    attn_kernel<<<512, 256, 0, stream>>>(Q, K, Vt, ctx);
    // K3: 128 * 16 = 2048 waves / 8 per block
    dense_kernel<<<256, 256, 0, stream>>>(ctx, wd, bd, out);
}